// SpectralConv_62929860821633
// MI455X (gfx1250) — compile-verified
//
#include <hip/hip_runtime.h>
#include <math.h>

// ---------------------------------------------------------------------------
// Spectral conv (FNO block) for MI455X / gfx1250, wave32, WMMA f32 16x16x4.
// All axis DFTs are expressed as rectangular GEMMs against constant DFT
// matrices (crop/fftshift folded into the frequency map), executed on
// v_wmma_f32_16x16x4_f32. Channel mix is weight-bandwidth-bound -> coalesced
// VALU streaming kernel.
// Workspace: two ping-pong complex buffers of 4*32*64*64*17 elems (2x71.3MB).
// Note: f32 WMMA has no A/B NEG modifier (NEG[1:0] must be 0), so the complex
// subtraction uses a register-negated B-imag fragment instead.
// ---------------------------------------------------------------------------

typedef __attribute__((ext_vector_type(2))) float v2f;
typedef __attribute__((ext_vector_type(8))) float v8f;

#define WMMA_P(A,B,C) __builtin_amdgcn_wmma_f32_16x16x4_f32(false,(A),false,(B),(short)0,(C),false,false)

#define NBATCH 4
#define NCIN   32
#define NCOUT  32
#define ND     64
#define NM     32
#define NE     17
#define NMODES (NM*NM*NE)   /* 17408 */
#define TWO_PI 6.28318530717958647692f

static __device__ __forceinline__ v2f mk2(float a, float b) { v2f r; r.x = a; r.y = b; return r; }

// ---------------------------------------------------------------------------
// K1: rfft along d3 with crop to 17 bins.
// Out[rows][17](complex) = X[rows][64](real) * E[64][17], E[n][k]=e^{-2pi i nk/64}
// One wave per 16 rows. rows = 4*32*64*64 = 524288.
// ---------------------------------------------------------------------------
__global__ __launch_bounds__(128) void k1_rfft_d3(const float* __restrict__ x,
                                                  v2f* __restrict__ out) {
  __shared__ v2f sBre[2][16][32];   // [ntile][kstep][lane], pre-swizzled B fragments
  __shared__ v2f sBim[2][16][32];
  const int tid = threadIdx.x;
  for (int i = tid; i < 2 * 16 * 32; i += 128) {
    int lane = i & 31, ks = (i >> 5) & 15, nt = i >> 9;
    int half = lane >> 4, l16 = lane & 15;
    int n = nt * 16 + l16;
    int k0 = ks * 4 + half * 2;
    float s0, c0, s1, c1;
    __sincosf(-TWO_PI * (float)(k0 * n) / 64.0f, &s0, &c0);
    __sincosf(-TWO_PI * (float)((k0 + 1) * n) / 64.0f, &s1, &c1);
    sBre[nt][ks][lane] = mk2(c0, c1);
    sBim[nt][ks][lane] = mk2(s0, s1);    // sin(-t) = -sin(t) = Im(E)
  }
  __syncthreads();

  const int lane = tid & 31, half = lane >> 4, l16 = lane & 15;
  const long wave = (long)((blockIdx.x * 128 + tid) >> 5);
  const long rowbase = wave * 16;
  const v2f* rowv = (const v2f*)(x + (rowbase + l16) * 64);

  v8f cr0 = {}, ci0 = {}, cr1 = {}, ci1 = {};
#pragma unroll
  for (int ks = 0; ks < 16; ++ks) {
    v2f a = rowv[ks * 2 + half];                 // A frag: M=l16, K=4ks+2*half+{0,1}
    cr0 = WMMA_P(a, sBre[0][ks][lane], cr0);
    ci0 = WMMA_P(a, sBim[0][ks][lane], ci0);
    cr1 = WMMA_P(a, sBre[1][ks][lane], cr1);
    ci1 = WMMA_P(a, sBim[1][ks][lane], ci1);
  }
#pragma unroll
  for (int j = 0; j < 8; ++j) {
    int m = j + half * 8;                        // C layout: lane half selects M+8
    long ob = (rowbase + m) * NE;
    out[ob + l16] = mk2(cr0[j], ci0[j]);         // ntile0: n = 0..15
    int n1 = 16 + l16;                           // ntile1: only n = 16 valid
    if (n1 < NE) out[ob + n1] = mk2(cr1[j], ci1[j]);
  }
}

// ---------------------------------------------------------------------------
// Generic complex DFT GEMM along d1/d2 (forward crop or inverse expand).
//   FWD : Out[m in 0..31][e] = sum_k e^{-2pi i f(m) k/64} In[k in 0..63][e]
//   INV : Out[m in 0..63][e] = (1/64) sum_k e^{+2pi i f(k) m/64} In[k in 0..31][e]
// with f(q) = (q+48)&63 (fftshift+center-crop folded in). N dim = e (17 of 32).
// One wave per GEMM; gemm id = gO*nInner + gI.
// ---------------------------------------------------------------------------
template <int MD, int KD, bool FWD>
__global__ __launch_bounds__(128) void kdft(const v2f* __restrict__ in, v2f* __restrict__ out,
                                            int nInner, long inO, long inI, long outO, long outI,
                                            int sK, int sM) {
  constexpr int MT = MD / 16, KS = KD / 4;
  __shared__ v2f sAre[MT][KS][32];   // pre-swizzled A fragments (constant DFT matrix)
  __shared__ v2f sAim[MT][KS][32];
  const int tid = threadIdx.x;
  for (int i = tid; i < MT * KS * 32; i += 128) {
    int lane = i & 31;
    int ks = (i >> 5) % KS;
    int mt = (i >> 5) / KS;
    int half = lane >> 4, l16 = lane & 15;
    int m = mt * 16 + l16;
    int k0 = ks * 4 + half * 2;
    float re[2], im[2];
#pragma unroll
    for (int q = 0; q < 2; ++q) {
      int k = k0 + q;
      float ang, scale;
      if (FWD) { int f = (m + 48) & 63; ang = -TWO_PI * (float)(f * k) / 64.0f; scale = 1.0f; }
      else     { int f = (k + 48) & 63; ang =  TWO_PI * (float)(f * m) / 64.0f; scale = 1.0f / 64.0f; }
      float s, c;
      __sincosf(ang, &s, &c);
      re[q] = c * scale;
      im[q] = s * scale;
    }
    sAre[mt][ks][lane] = mk2(re[0], re[1]);
    sAim[mt][ks][lane] = mk2(im[0], im[1]);
  }
  __syncthreads();

  const int lane = tid & 31, half = lane >> 4, l16 = lane & 15;
  const int wave = (blockIdx.x * 128 + tid) >> 5;
  const int gO = wave / nInner;
  const int gI = wave % nInner;
  const v2f* bin = in + gO * inO + gI * inI;
  v2f* bout = out + gO * outO + gI * outI;
  const int e0 = l16;
  const int e1 = 16 + l16;
  const v2f z = mk2(0.0f, 0.0f);

  v8f accR[MT][2] = {}, accI[MT][2] = {};
  for (int ks = 0; ks < KS; ++ks) {
    int k0 = ks * 4 + half * 2;
    v2f b0n0 = bin[(long)(k0    ) * sK + e0];
    v2f b1n0 = bin[(long)(k0 + 1) * sK + e0];
    v2f b0n1 = (e1 < NE) ? bin[(long)(k0    ) * sK + e1] : z;
    v2f b1n1 = (e1 < NE) ? bin[(long)(k0 + 1) * sK + e1] : z;
    v2f brn0 = mk2(b0n0.x, b1n0.x), bim0 = mk2(b0n0.y, b1n0.y);
    v2f brn1 = mk2(b0n1.x, b1n1.x), bim1 = mk2(b0n1.y, b1n1.y);
    v2f bim0n = mk2(-bim0.x, -bim0.y);           // f32 WMMA has no neg_a/neg_b:
    v2f bim1n = mk2(-bim1.x, -bim1.y);           // negate B-imag in registers
#pragma unroll
    for (int mt = 0; mt < MT; ++mt) {
      v2f ar = sAre[mt][ks][lane];
      v2f ai = sAim[mt][ks][lane];
      // complex GEMM: Cr += Ar*Br + Ai*(-Bi), Ci += Ar*Bi + Ai*Br
      accR[mt][0] = WMMA_P(ar, brn0,  accR[mt][0]);
      accR[mt][0] = WMMA_P(ai, bim0n, accR[mt][0]);
      accI[mt][0] = WMMA_P(ar, bim0,  accI[mt][0]);
      accI[mt][0] = WMMA_P(ai, brn0,  accI[mt][0]);
      accR[mt][1] = WMMA_P(ar, brn1,  accR[mt][1]);
      accR[mt][1] = WMMA_P(ai, bim1n, accR[mt][1]);
      accI[mt][1] = WMMA_P(ar, bim1,  accI[mt][1]);
      accI[mt][1] = WMMA_P(ai, brn1,  accI[mt][1]);
    }
  }
#pragma unroll
  for (int mt = 0; mt < MT; ++mt)
#pragma unroll
    for (int j = 0; j < 8; ++j) {
      int m = mt * 16 + j + half * 8;
      bout[(long)m * sM + e0] = mk2(accR[mt][0][j], accI[mt][0][j]);
      if (e1 < NE) bout[(long)m * sM + e1] = mk2(accR[mt][1][j], accI[mt][1][j]);
    }
}

// ---------------------------------------------------------------------------
// K4: per-mode complex channel mix: out[a,f,m] = sum_b x[a,b,m] * (wr+i wi)[b,f,m]
// Weight-bandwidth-bound (4 flop/byte): one lane per mode (coalesced along the
// fastest-varying mode dim), weights streamed exactly once (f chunked into two
// register-resident halves that partition the weight tensor).
// ---------------------------------------------------------------------------
__global__ __launch_bounds__(256) void k4_mix(const v2f* __restrict__ xs,
                                              const float* __restrict__ wr,
                                              const float* __restrict__ wi,
                                              v2f* __restrict__ out) {
  const int m = blockIdx.x * 256 + threadIdx.x;   // grid sized exactly to NMODES
#pragma unroll
  for (int fc = 0; fc < 2; ++fc) {
    float ar[4][16], ai[4][16];
#pragma unroll
    for (int a = 0; a < 4; ++a)
#pragma unroll
      for (int f = 0; f < 16; ++f) { ar[a][f] = 0.0f; ai[a][f] = 0.0f; }
    for (int b = 0; b < NCIN; ++b) {
      v2f xv[4];
#pragma unroll
      for (int a = 0; a < 4; ++a) xv[a] = xs[(long)(a * NCIN + b) * NMODES + m];
#pragma unroll
      for (int f = 0; f < 16; ++f) {
        long wofs = (long)(b * NCOUT + fc * 16 + f) * NMODES + m;
        float wrv = wr[wofs], wiv = wi[wofs];
#pragma unroll
        for (int a = 0; a < 4; ++a) {
          ar[a][f] += xv[a].x * wrv - xv[a].y * wiv;
          ai[a][f] += xv[a].x * wiv + xv[a].y * wrv;
        }
      }
    }
#pragma unroll
    for (int a = 0; a < 4; ++a)
#pragma unroll
      for (int f = 0; f < 16; ++f)
        out[(long)(a * NCOUT + fc * 16 + f) * NMODES + m] = mk2(ar[a][f], ai[a][f]);
  }
}

// ---------------------------------------------------------------------------
// K7: irfft along d3 (17 complex -> 64 real) + bias, as a real GEMM with the
// re/im parts stacked along K (K = 34, padded to 36):
//   out[r][n] = sum_{kk} A[r][kk] * E[kk][n],  A = [Re X | Im X]
//   E[k][n] = c_k cos(2pi kn/64),  E[17+k][n] = -c_k sin(2pi kn/64),
//   c_0 = 1/64, c_k = 2/64 (Im of DC naturally dropped since sin(0)=0).
// ---------------------------------------------------------------------------
__global__ __launch_bounds__(128) void k7_irfft_d3(const v2f* __restrict__ s6,
                                                   const float* __restrict__ bias,
                                                   float* __restrict__ out) {
  __shared__ v2f sB[4][9][32];   // [ntile][kstep][lane]
  const int tid = threadIdx.x;
  for (int i = tid; i < 4 * 9 * 32; i += 128) {
    int lane = i & 31;
    int ks = (i >> 5) % 9;
    int nt = (i >> 5) / 9;
    int half = lane >> 4, l16 = lane & 15;
    int n = nt * 16 + l16;
    float v[2];
#pragma unroll
    for (int q = 0; q < 2; ++q) {
      int kk = ks * 4 + half * 2 + q;
      float val = 0.0f;
      if (kk < 17) {
        float coef = (kk == 0 ? 1.0f : 2.0f) / 64.0f;
        val = coef * __cosf(TWO_PI * (float)(kk * n) / 64.0f);
      } else if (kk < 34) {
        int k = kk - 17;
        float coef = (k == 0 ? 1.0f : 2.0f) / 64.0f;
        val = -coef * __sinf(TWO_PI * (float)(k * n) / 64.0f);
      }
      v[q] = val;
    }
    sB[nt][ks][lane] = mk2(v[0], v[1]);
  }
  __syncthreads();

  const int lane = tid & 31, half = lane >> 4, l16 = lane & 15;
  const long wave = (long)((blockIdx.x * 128 + tid) >> 5);
  const long rowbase = wave * 16;
  const v2f* rowp = s6 + (rowbase + l16) * NE;
  const float bv = bias[(int)((rowbase >> 12) & 31)];   // 16 rows share (b,f,d1)

  v8f acc[4] = {};
#pragma unroll
  for (int ks = 0; ks < 9; ++ks) {
    int k0 = ks * 4 + half * 2;
    int k1 = k0 + 1;
    float a0 = (k0 < 17) ? rowp[k0].x : ((k0 < 34) ? rowp[k0 - 17].y : 0.0f);
    float a1 = (k1 < 17) ? rowp[k1].x : ((k1 < 34) ? rowp[k1 - 17].y : 0.0f);
    v2f a = mk2(a0, a1);
#pragma unroll
    for (int nt = 0; nt < 4; ++nt)
      acc[nt] = WMMA_P(a, sB[nt][ks][lane], acc[nt]);
  }
#pragma unroll
  for (int nt = 0; nt < 4; ++nt)
#pragma unroll
    for (int j = 0; j < 8; ++j) {
      int m = j + half * 8;
      out[(rowbase + m) * 64 + nt * 16 + l16] = acc[nt][j] + bv;
    }
}

// ---------------------------------------------------------------------------
extern "C" void kernel_launch(void* const* d_in, const int* in_sizes, int n_in,
                              void* d_out, int out_size, void* d_ws, size_t ws_size,
                              hipStream_t stream) {
  const float* x    = (const float*)d_in[0];
  const float* wr   = (const float*)d_in[1];
  const float* wi   = (const float*)d_in[2];
  const float* bias = (const float*)d_in[3];
  float* out = (float*)d_out;

  const long NA = (long)NBATCH * NCIN * ND * ND * NE;   // 8,912,896 complex
  v2f* bufA = (v2f*)d_ws;
  v2f* bufB = bufA + NA;                                // needs 2*71.3MB of d_ws

  // K1: d3 rfft+crop: [b][c][d1][d2][64]f32 -> bufA [b][c][d1][d2][17]c
  k1_rfft_d3<<<dim3(8192), dim3(128), 0, stream>>>(x, bufA);

  // K2: fwd DFT d2 (64->32): bufA -> bufB [b][c][d1][m2][17]; gemm per (b,c,d1)
  kdft<32, 64, true><<<dim3(2048), dim3(128), 0, stream>>>(
      bufA, bufB, 1, (long)ND * NE, 0L, (long)NM * NE, 0L, NE, NE);

  // K3: fwd DFT d1 (64->32): bufB -> bufA [b][c][m1][m2][17]; gemm per (b,c)x(m2)
  kdft<32, 64, true><<<dim3(1024), dim3(128), 0, stream>>>(
      bufB, bufA, 32, (long)ND * NM * NE, (long)NE, (long)NM * NM * NE, (long)NE,
      NM * NE, NM * NE);

  // K4: channel mix: bufA [a][b][mode] -> bufB [a][f][mode]
  k4_mix<<<dim3(NMODES / 256), dim3(256), 0, stream>>>(bufA, wr, wi, bufB);

  // K5: inv DFT d1 (32->64): bufB -> bufA [b][f][d1][m2][17]; gemm per (b,f)x(m2)
  kdft<64, 32, false><<<dim3(1024), dim3(128), 0, stream>>>(
      bufB, bufA, 32, (long)NM * NM * NE, (long)NE, (long)ND * NM * NE, (long)NE,
      NM * NE, NM * NE);

  // K6: inv DFT d2 (32->64): bufA -> bufB [b][f][d1][d2][17]; gemm per (b,f,d1)
  kdft<64, 32, false><<<dim3(2048), dim3(128), 0, stream>>>(
      bufA, bufB, 1, (long)NM * NE, 0L, (long)ND * NE, 0L, NE, NE);

  // K7: d3 irfft + bias: bufB -> out [b][f][d1][d2][64]f32
  k7_irfft_d3<<<dim3(8192), dim3(128), 0, stream>>>(bufB, bias, out);
}